// EaRL_1735166788691
// MI455X (gfx1250) — compile-verified
//
#include <hip/hip_runtime.h>
#include <hip/hip_bf16.h>

// ---------------------------------------------------------------------------
// Heterogeneous GAT (4 layers, 19 edge types) + gene MLP heads for gfx1250.
// Dense GEMMs use V_WMMA_F32_16X16X4_F32 (fp32 matrix cores, wave32).
// Edge softmax uses global atomic max (ordered-uint trick) + atomic f32 adds.
// ---------------------------------------------------------------------------

typedef __attribute__((ext_vector_type(2))) float v2f;
typedef __attribute__((ext_vector_type(8))) float v8f;

#define NEG_SLOPE 0.2f

__device__ __forceinline__ unsigned fkey(float f) {
  unsigned b = __float_as_uint(f);
  return (b & 0x80000000u) ? ~b : (b | 0x80000000u);
}
__device__ __forceinline__ float fdec(unsigned k) {
  if (k == 0u) return 0.0f;  // untouched -> isolated dst -> amax = 0 (ref isfinite path)
  unsigned b = (k & 0x80000000u) ? (k & 0x7fffffffu) : ~k;
  return __uint_as_float(b);
}

__global__ void zero_buf(float* __restrict__ p, long n) {
  long i = (long)blockIdx.x * blockDim.x + threadIdx.x;
  long stride = (long)gridDim.x * blockDim.x;
  for (; i < n; i += stride) p[i] = 0.0f;
}

// out[n,h] = sum_f x[n,f] * wv[f,h], wv[f,h] = sum_c W[f, h*32+c]*att[h,c]
__global__ void att_proj(const float* __restrict__ x, int ldx,
                         const float* __restrict__ W,
                         const float* __restrict__ att,
                         float* __restrict__ out, int N, int fin) {
  __shared__ float wv[128];  // fin*2, fin <= 64
  for (int idx = threadIdx.x; idx < fin * 2; idx += blockDim.x) {
    int f = idx >> 1, h = idx & 1;
    float s = 0.f;
    for (int c = 0; c < 32; ++c) s += W[f * 64 + h * 32 + c] * att[h * 32 + c];
    wv[idx] = s;
  }
  __syncthreads();
  int n = blockIdx.x * blockDim.x + threadIdx.x;
  if (n >= N) return;
  const float* xr = x + (size_t)n * ldx;
  float a0 = 0.f, a1 = 0.f;
  for (int f = 0; f < fin; ++f) {
    float xv = xr[f];
    a0 += xv * wv[f * 2];
    a1 += xv * wv[f * 2 + 1];
  }
  out[n * 2] = a0;
  out[n * 2 + 1] = a1;
}

// D[M,N] = A[M,K] @ B[K,N] (+bias, optional relu). One wave computes a 16x64
// tile via 4x V_WMMA_F32_16X16X4_F32 accumulators. K % 4 == 0; N-tile of 64
// selected by blockIdx.y; 4 waves/block cover 4 M-tiles.
// fp32 WMMA 16x16x4 VGPR layout (ISA 7.12.2):
//   A 16x4: lanes 0-15 rows M=lane hold K,K+1 in v0,v1; lanes 16-31 hold K+2,K+3
//   B 4x16: lanes 0-15 col N=lane rows K,K+1; lanes 16-31 rows K+2,K+3
//   C/D 16x16: VGPR i = row i (lanes 0-15) / row 8+i (lanes 16-31), col = lane&15
// OOB M rows: A address is clamped to a valid row (row m of A only feeds row m
// of D, and stores are guarded), so the inner loop has no predication at all.
__global__ void wmma_gemm_f32(const float* __restrict__ A, int lda,
                              const float* __restrict__ B, int ldb,
                              float* __restrict__ D, int ldd,
                              int M, int K,
                              const float* __restrict__ bias, int relu) {
  int wave = threadIdx.x >> 5;
  int lane = threadIdx.x & 31;
  int m0 = (blockIdx.x * 4 + wave) * 16;
  if (m0 >= M) return;  // wave-uniform exit; EXEC stays all-ones for WMMA
  int n0 = blockIdx.y * 64;

  v8f acc[4] = {};
  int arow = m0 + (lane & 15);
  if (arow > M - 1) arow = M - 1;       // clamp: keeps loads in-bounds, rows >= M discarded
  int akoff = (lane < 16) ? 0 : 2;
  const float* Arow = A + (size_t)arow * lda + akoff;          // 8B-aligned at every k
  const float* Bp = B + (size_t)akoff * ldb + n0 + (lane & 15);

  for (int k = 0; k < K; k += 4) {
    v2f a = *(const v2f*)(Arow + k);                           // single b64 load
    const float* B0 = Bp + (size_t)k * ldb;
#pragma unroll
    for (int t = 0; t < 4; ++t) {
      v2f b;
      b.x = B0[t * 16];
      b.y = B0[(size_t)ldb + t * 16];
      acc[t] = __builtin_amdgcn_wmma_f32_16x16x4_f32(
          false, a, false, b, (short)0, acc[t], false, false);
    }
  }

  int rbase = m0 + ((lane < 16) ? 0 : 8);
  int bcol0 = lane & 15;
#pragma unroll
  for (int t = 0; t < 4; ++t) {
    int col = n0 + t * 16 + bcol0;
    float bv = bias ? bias[col] : 0.f;
#pragma unroll
    for (int i = 0; i < 8; ++i) {
      int row = rbase + i;
      if (row < M) {
        float v = acc[t][i] + bv;
        if (relu) v = fmaxf(v, 0.f);
        D[(size_t)row * ldd + col] = v;
      }
    }
  }
}

__global__ void edge_alpha_max(const int* __restrict__ src, const int* __restrict__ dst,
                               const float* __restrict__ as_, const float* __restrict__ ad_,
                               float* __restrict__ alpha, unsigned* __restrict__ amax, int E) {
  int e = blockIdx.x * blockDim.x + threadIdx.x;
  if (e >= E) return;
  int s = src[e], d = dst[e];
#pragma unroll
  for (int h = 0; h < 2; ++h) {
    float al = as_[s * 2 + h] + ad_[d * 2 + h];
    al = (al > 0.f) ? al : NEG_SLOPE * al;  // leaky_relu
    alpha[e * 2 + h] = al;
    atomicMax(&amax[d * 2 + h], fkey(al));
  }
}

__global__ void edge_exp_sum(const int* __restrict__ dst,
                             float* __restrict__ alpha, const unsigned* __restrict__ amax,
                             float* __restrict__ denom, int E) {
  int e = blockIdx.x * blockDim.x + threadIdx.x;
  if (e >= E) return;
  int d = dst[e];
#pragma unroll
  for (int h = 0; h < 2; ++h) {
    float mx = fdec(amax[d * 2 + h]);
    float ex = __expf(alpha[e * 2 + h] - mx);
    alpha[e * 2 + h] = ex;  // reuse buffer for ex
    atomicAdd(&denom[d * 2 + h], ex);
  }
}

// thread t -> (edge, head); 32 atomic f32 adds of w * hs[src] into acc[dst]
__global__ void edge_aggregate(const int* __restrict__ src, const int* __restrict__ dst,
                               const float* __restrict__ ex, const float* __restrict__ denom,
                               const float* __restrict__ hs, float* __restrict__ acc, int E) {
  int t = blockIdx.x * blockDim.x + threadIdx.x;
  if (t >= 2 * E) return;
  int e = t >> 1, h = t & 1;
  int s = src[e], d = dst[e];
  float w = ex[e * 2 + h] / denom[d * 2 + h];
  const float4* hrow = (const float4*)(hs + (size_t)s * 64 + h * 32);
  float* arow = acc + (size_t)d * 64 + h * 32;
#pragma unroll
  for (int q = 0; q < 8; ++q) {
    float4 v = hrow[q];
    atomicAdd(&arow[q * 4 + 0], v.x * w);
    atomicAdd(&arow[q * 4 + 1], v.y * w);
    atomicAdd(&arow[q * 4 + 2], v.z * w);
    atomicAdd(&arow[q * 4 + 3], v.w * w);
  }
}

__global__ void bias_accum(const float* __restrict__ b, float* __restrict__ bacc) {
  int c = threadIdx.x;
  if (c < 32) bacc[c] += b[c];
}

// x_new[n,c] = relu(0.5*(acc[n,0,c]+acc[n,1,c]) + bias_sum[c])
__global__ void finalize_nodes(const float* __restrict__ acc, const float* __restrict__ bacc,
                               float* __restrict__ xout, int N) {
  int t = blockIdx.x * blockDim.x + threadIdx.x;
  if (t >= N * 32) return;
  int n = t >> 5, c = t & 31;
  float v = 0.5f * (acc[(size_t)n * 64 + c] + acc[(size_t)n * 64 + 32 + c]) + bacc[c];
  xout[(size_t)n * 32 + c] = fmaxf(v, 0.f);
}

// final 512->1 dots for both heads + sigmoid + bernoulli threshold
__global__ void head_final(const float* __restrict__ t2v, const float* __restrict__ Wv,
                           const float* __restrict__ bv,
                           const float* __restrict__ t2z, const float* __restrict__ Wz,
                           const float* __restrict__ bz,
                           const float* __restrict__ u, float* __restrict__ out, int N) {
  int wid = (int)((blockIdx.x * blockDim.x + threadIdx.x) >> 5);
  int lane = threadIdx.x & 31;
  if (wid >= N) return;  // wave-uniform
  const float* rv = t2v + (size_t)wid * 512;
  const float* rz = t2z + (size_t)wid * 512;
  float sv = 0.f, sz = 0.f;
  for (int i = lane; i < 512; i += 32) {
    sv += rv[i] * Wv[i];
    sz += rz[i] * Wz[i];
  }
#pragma unroll
  for (int offd = 16; offd > 0; offd >>= 1) {
    sv += __shfl_down(sv, offd, 32);
    sz += __shfl_down(sz, offd, 32);
  }
  if (lane == 0) {
    float val = sv + bv[0];
    float p = 1.f / (1.f + __expf(-(sz + bz[0])));
    out[wid] = val;
    out[N + wid] = p;
    out[2 * N + wid] = (u[wid] < p) ? 1.f : 0.f;
  }
}

// ---------------------------------------------------------------------------
extern "C" void kernel_launch(void* const* d_in, const int* in_sizes, int n_in,
                              void* d_out, int out_size, void* d_ws, size_t ws_size,
                              hipStream_t stream) {
  // node types: 0 tad, 1 atac_region, 2 gene, 3 protein, 4 protein_name, 5 enhancer
  const int NODE_N[6]   = {5000, 200000, 30000, 15000, 15000, 50000};
  const int NODE_OFF[6] = {0, 5000, 205000, 235000, 250000, 265000};
  const int TOT = 315000;
  const int ET_SRC[19] = {0,1,0,2,1,2,3,3,2,3,2,3,4,5,1,5,2,1,2};
  const int ET_DST[19] = {1,0,2,0,2,1,3,2,3,2,3,4,3,1,5,2,5,2,1};
  const int NE = 19;
  const int E = in_sizes[6] / 2;  // edge_index tensors are [2, E]

  // input layout (setup_inputs insertion order):
  //  0..5   x_dict (tad, atac, gene, protein, protein_name, enhancer) [N,64] f32
  //  6..24  edge_index per EDGE_TYPES, [2,E] i32
  //  25 + (L*19+e)*5 + {Ws,Wd,att_s,att_d,b}
  //  405..410 gene_zero (W0,b0,W1,b1,W2,b2); 411..416 gene_value; 417 u_noise
  (void)n_in; (void)out_size; (void)ws_size;

  size_t off = 0;
  auto carve = [&](size_t bytes) -> char* {
    char* p = (char*)d_ws + off;
    off += (bytes + 255) & ~(size_t)255;
    return p;
  };
  float*    xA    = (float*)carve((size_t)TOT * 32 * 4);
  float*    xB    = (float*)carve((size_t)TOT * 32 * 4);
  float*    acc   = (float*)carve((size_t)TOT * 64 * 4);
  float*    hs    = (float*)carve((size_t)200000 * 64 * 4);
  float*    a_s   = (float*)carve((size_t)200000 * 2 * 4);
  float*    a_d   = (float*)carve((size_t)200000 * 2 * 4);
  float*    alpha = (float*)carve((size_t)E * 2 * 4);
  unsigned* amax  = (unsigned*)carve((size_t)200000 * 2 * 4);
  float*    denom = (float*)carve((size_t)200000 * 2 * 4);
  float*    bacc  = (float*)carve(6 * 32 * 4);
  float*    t1    = (float*)carve((size_t)30000 * 512 * 4);
  float*    t2v   = (float*)carve((size_t)30000 * 512 * 4);
  float*    t2z   = (float*)carve((size_t)30000 * 512 * 4);

  for (int L = 0; L < 4; ++L) {
    int fin = (L == 0) ? 64 : 32;
    float* xout = (L % 2 == 0) ? xA : xB;
    const float* xprev = (L % 2 == 0) ? xB : xA;

    zero_buf<<<2048, 256, 0, stream>>>(acc, (long)TOT * 64);
    zero_buf<<<1, 256, 0, stream>>>(bacc, 6 * 32);

    for (int e = 0; e < NE; ++e) {
      int st = ET_SRC[e], dt = ET_DST[e];
      const float* xs; const float* xd; int ld;
      if (L == 0) { xs = (const float*)d_in[st]; xd = (const float*)d_in[dt]; ld = 64; }
      else { xs = xprev + (size_t)NODE_OFF[st] * 32; xd = xprev + (size_t)NODE_OFF[dt] * 32; ld = 32; }
      int Ns = NODE_N[st], Nd = NODE_N[dt];
      int pb = 25 + (L * 19 + e) * 5;
      const float* Ws   = (const float*)d_in[pb + 0];
      const float* Wd   = (const float*)d_in[pb + 1];
      const float* atts = (const float*)d_in[pb + 2];
      const float* attd = (const float*)d_in[pb + 3];
      const float* bvec = (const float*)d_in[pb + 4];
      const int* ei = (const int*)d_in[6 + e];
      const int* srcI = ei;
      const int* dstI = ei + E;

      att_proj<<<(Ns + 255) / 256, 256, 0, stream>>>(xs, ld, Ws, atts, a_s, Ns, fin);
      att_proj<<<(Nd + 255) / 256, 256, 0, stream>>>(xd, ld, Wd, attd, a_d, Nd, fin);

      int mtiles = (Ns + 15) / 16;
      wmma_gemm_f32<<<dim3((mtiles + 3) / 4, 1), 128, 0, stream>>>(
          xs, ld, Ws, 64, hs, 64, Ns, fin, nullptr, 0);

      zero_buf<<<(2 * Nd + 255) / 256, 256, 0, stream>>>((float*)amax, 2 * Nd);
      zero_buf<<<(2 * Nd + 255) / 256, 256, 0, stream>>>(denom, 2 * Nd);
      edge_alpha_max<<<(E + 255) / 256, 256, 0, stream>>>(srcI, dstI, a_s, a_d, alpha, amax, E);
      edge_exp_sum<<<(E + 255) / 256, 256, 0, stream>>>(dstI, alpha, amax, denom, E);
      edge_aggregate<<<(2 * E + 255) / 256, 256, 0, stream>>>(
          srcI, dstI, alpha, denom, hs, acc + (size_t)NODE_OFF[dt] * 64, E);
      bias_accum<<<1, 32, 0, stream>>>(bvec, bacc + dt * 32);
    }
    for (int t = 0; t < 6; ++t) {
      long nthr = (long)NODE_N[t] * 32;
      finalize_nodes<<<(nthr + 255) / 256, 256, 0, stream>>>(
          acc + (size_t)NODE_OFF[t] * 64, bacc + t * 32,
          xout + (size_t)NODE_OFF[t] * 32, NODE_N[t]);
    }
  }

  // gene head: features live in xB after layer 3
  const float* hgene = xB + (size_t)NODE_OFF[2] * 32;
  int NG = NODE_N[2];
  const float* zW0 = (const float*)d_in[405]; const float* zb0 = (const float*)d_in[406];
  const float* zW1 = (const float*)d_in[407]; const float* zb1 = (const float*)d_in[408];
  const float* zW2 = (const float*)d_in[409]; const float* zb2 = (const float*)d_in[410];
  const float* vW0 = (const float*)d_in[411]; const float* vb0 = (const float*)d_in[412];
  const float* vW1 = (const float*)d_in[413]; const float* vb1 = (const float*)d_in[414];
  const float* vW2 = (const float*)d_in[415]; const float* vb2 = (const float*)d_in[416];
  const float* unoise = (const float*)d_in[417];

  int gmt = ((NG + 15) / 16 + 3) / 4;
  // value head (layers 1-2 with relu)
  wmma_gemm_f32<<<dim3(gmt, 8), 128, 0, stream>>>(hgene, 32, vW0, 512, t1, 512, NG, 32, vb0, 1);
  wmma_gemm_f32<<<dim3(gmt, 8), 128, 0, stream>>>(t1, 512, vW1, 512, t2v, 512, NG, 512, vb1, 1);
  // zero head
  wmma_gemm_f32<<<dim3(gmt, 8), 128, 0, stream>>>(hgene, 32, zW0, 512, t1, 512, NG, 32, zb0, 1);
  wmma_gemm_f32<<<dim3(gmt, 8), 128, 0, stream>>>(t1, 512, zW1, 512, t2z, 512, NG, 512, zb1, 1);
  // out = [values | p_zero | zeros]
  head_final<<<((long)NG * 32 + 127) / 128, 128, 0, stream>>>(
      t2v, vW2, vb2, t2z, zW2, zb2, unoise, (float*)d_out, NG);
}